// CRF_24361054503468
// MI455X (gfx1250) — compile-verified
//
#include <hip/hip_runtime.h>

// CRF forward (log-partition) on gfx1250.
// Exp-domain reformulation: alpha' = u_t + m + log( E @ exp(alpha - m) ),
// E = exp(trans) held in registers as V_WMMA_F32_16X16X4_F32 A-fragments.
// Block = 4 waves (128 thr, wave32); block owns 16 batch columns (N dim),
// wave w owns tag tile [16w,16w+16) (M dim); K = 64 tags = 16 k-steps.
//
// Round-2 changes: (a) stage all 16 B-fragments into registers before the
// WMMA phase so ds_loads pipeline instead of wait-per-wmma, (b) dual
// accumulator chains so the matrix pipe overlaps independent WMMAs,
// (c) hoist the unary b128 loads to the top of the iteration to hide
// global latency behind the reduction/exp/LDS/WMMA work.

#define NTAGS 64
#define TLEN  512
#define NEGV  (-10000.0f)
#define PAD   20   // LDS row stride (floats): 8*PAD % 64 == 32 -> halves hit disjoint banks

typedef float v2f __attribute__((ext_vector_type(2)));
typedef float v8f __attribute__((ext_vector_type(8)));

__launch_bounds__(128, 1)
__global__ void crf_fwd_wmma(const float* __restrict__ unary,
                             const float* __restrict__ trans,
                             const long long* __restrict__ lengths,
                             float* __restrict__ out) {
  __shared__ float A_lds[NTAGS * PAD]; // exp-domain state: [tag][batch-col]
  __shared__ float Mx[64];             // per-wave per-column maxes
  __shared__ float Sx[64];             // per-wave per-column partial sums

  const int tid   = threadIdx.x;
  const int w     = tid >> 5;          // wave id 0..3 -> M tile
  const int lane  = tid & 31;
  const int half  = lane >> 4;         // lane half selects K sub-cols per ISA layout
  const int col   = lane & 15;         // batch column within tile / M row for A frags
  const int wbase = w * 16;
  const int b0    = blockIdx.x * 16;
  const int myb   = b0 + col;

  // ---- Preload E = exp(trans) as A-fragments (loop-invariant, 32 VGPRs) ----
  // A 16x4 layout: VGPR0 = K=4k+2*half, VGPR1 = K=4k+2*half+1, M = col.
  float eA0[16], eA1[16];
  {
    const float* tp = trans + (wbase + col) * NTAGS + 2 * half;
#pragma unroll
    for (int k = 0; k < 16; ++k) {
      eA0[k] = __expf(tp[4 * k + 0]);
      eA1[k] = __expf(tp[4 * k + 1]);
    }
  }

  const long long mylen = lengths[myb];

  // alphas resident in WMMA C/D layout: al[r] = alpha[wbase + r + 8*half][col]
  float al[8];
#pragma unroll
  for (int r = 0; r < 8; ++r) {
    const int tag = wbase + r + 8 * half;
    al[r] = (tag == 1) ? 0.0f : NEGV;  // START_IDX = 1
  }

  const float* ubase = unary + (size_t)myb * TLEN * NTAGS + wbase + 8 * half;

  for (int t = 0; t < TLEN; ++t) {
    // ---- issue emission loads early (consumed at end of iteration) ----
    const float* up = ubase + (size_t)t * NTAGS;
    const float4 u0 = *(const float4*)(up);
    const float4 u1 = *(const float4*)(up + 4);
    if (t + 1 < TLEN) __builtin_prefetch(up + NTAGS, 0, 3);  // global_prefetch_b8

    // ---- global per-column max (numerical stability) ----
    float pm = al[0];
#pragma unroll
    for (int r = 1; r < 8; ++r) pm = fmaxf(pm, al[r]);
    pm = fmaxf(pm, __shfl_xor(pm, 16, 32));
    if (lane < 16) Mx[wbase + col] = pm;
    __syncthreads();  // also fences prior iteration's A_lds reads
    const float m = fmaxf(fmaxf(Mx[col], Mx[16 + col]),
                          fmaxf(Mx[32 + col], Mx[48 + col]));

    // ---- exp-domain state vector -> LDS ----
#pragma unroll
    for (int r = 0; r < 8; ++r)
      A_lds[(wbase + r + 8 * half) * PAD + col] = __expf(al[r] - m);
    __syncthreads();

    // ---- stage all B-fragments (K x 16 columns) into registers ----
    // B 4x16 layout: VGPR v = row v (lanes 0-15) | row v+2 (lanes 16-31).
    float bf0[16], bf1[16];
#pragma unroll
    for (int k = 0; k < 16; ++k) {
      const int j0 = 4 * k + 2 * half;
      bf0[k] = A_lds[j0 * PAD + col];
      bf1[k] = A_lds[(j0 + 1) * PAD + col];
    }

    // ---- y = E_tile @ a  via 16 x V_WMMA_F32_16X16X4_F32 (K = 64),
    //      two independent accumulation chains ----
    v8f acc0 = {0.f, 0.f, 0.f, 0.f, 0.f, 0.f, 0.f, 0.f};
    v8f acc1 = {0.f, 0.f, 0.f, 0.f, 0.f, 0.f, 0.f, 0.f};
#pragma unroll
    for (int k = 0; k < 16; k += 2) {
      v2f Av0, Bv0, Av1, Bv1;
      Av0[0] = eA0[k];     Av0[1] = eA1[k];
      Bv0[0] = bf0[k];     Bv0[1] = bf1[k];
      Av1[0] = eA0[k + 1]; Av1[1] = eA1[k + 1];
      Bv1[0] = bf0[k + 1]; Bv1[1] = bf1[k + 1];
      acc0 = __builtin_amdgcn_wmma_f32_16x16x4_f32(
          false, Av0, false, Bv0, (short)0, acc0, false, false);
      acc1 = __builtin_amdgcn_wmma_f32_16x16x4_f32(
          false, Av1, false, Bv1, (short)0, acc1, false, false);
    }

    // ---- back to log domain, length mask ----
    const float un[8] = {u0.x, u0.y, u0.z, u0.w, u1.x, u1.y, u1.z, u1.w};
    const bool keep = ((long long)t < mylen);
#pragma unroll
    for (int r = 0; r < 8; ++r) {
      const float nv = un[r] + m + __logf(acc0[r] + acc1[r]);
      al[r] = keep ? nv : al[r];
    }
  }

  // ---- terminal: alpha += trans[END=2][tag]; logsumexp over 64 tags ----
  {
    const float* tr2 = trans + 2 * NTAGS + wbase + 8 * half;
    const float4 e0 = *(const float4*)(tr2);
    const float4 e1 = *(const float4*)(tr2 + 4);
    const float te[8] = {e0.x, e0.y, e0.z, e0.w, e1.x, e1.y, e1.z, e1.w};
#pragma unroll
    for (int r = 0; r < 8; ++r) al[r] += te[r];
  }
  float pm = al[0];
#pragma unroll
  for (int r = 1; r < 8; ++r) pm = fmaxf(pm, al[r]);
  pm = fmaxf(pm, __shfl_xor(pm, 16, 32));
  if (lane < 16) Mx[wbase + col] = pm;
  __syncthreads();
  const float m = fmaxf(fmaxf(Mx[col], Mx[16 + col]),
                        fmaxf(Mx[32 + col], Mx[48 + col]));
  float s = 0.f;
#pragma unroll
  for (int r = 0; r < 8; ++r) s += __expf(al[r] - m);
  s += __shfl_xor(s, 16, 32);
  if (lane < 16) Sx[wbase + col] = s;
  __syncthreads();
  if (w == 0 && lane < 16)
    out[b0 + col] =
        m + __logf(Sx[col] + Sx[16 + col] + Sx[32 + col] + Sx[48 + col]);
}

extern "C" void kernel_launch(void* const* d_in, const int* in_sizes, int n_in,
                              void* d_out, int out_size, void* d_ws, size_t ws_size,
                              hipStream_t stream) {
  const float*     unary   = (const float*)d_in[0];     // [B, T, 64] f32
  const float*     trans   = (const float*)d_in[1];     // [1, 64, 64] f32
  const long long* lengths = (const long long*)d_in[2]; // [B] i64
  float*           out     = (float*)d_out;             // [B] f32

  const int B = in_sizes[2];          // 1024
  const int grid = B / 16;            // 16 batch columns per block
  crf_fwd_wmma<<<grid, 128, 0, stream>>>(unary, trans, lengths, out);
}